// VolumeRender_43877385896408
// MI455X (gfx1250) — compile-verified
//
#include <hip/hip_runtime.h>
#include <math.h>
#include <stdint.h>

#define D_DISP  192
#define H_DIM   192
#define W_DIM   384
#define WV_DIM  575          // W + D - 1
#define TPB     128          // 4 wave32s per block
#define PIPE    4            // async pipeline depth (circular buffers)

// LDS staging: [PIPE buffers][8 slots: ch0..3 path1, ch0..3 path2][TPB lanes]
// 4*8*128*4 = 16 KB. Lane i writes and later reads only its own slots, so the
// only synchronization needed is the per-wave ASYNCcnt (+ a DScnt WAR fence).

__global__ __launch_bounds__(TPB) void volrender_kernel(
    const float* __restrict__ v, float* __restrict__ out)
{
    const int tid = threadIdx.x;
    const int w   = blockIdx.x * TPB + tid;   // 3 blocks cover W=384 exactly
    const int h   = blockIdx.y;

    __shared__ float sm[PIPE][8][TPB];

    const uint32_t stride_d  = (uint32_t)H_DIM * WV_DIM;   // 110400 elems
    const uint32_t stride_ch = (uint32_t)D_DISP * stride_d;
    const uint32_t base      = (uint32_t)h * WV_DIM + (uint32_t)w;

    // Issue 8 async global->LDS loads for disparity slice d into buffer bb.
    // GVS mode: mem_addr = SGPR64(base ptr) + VGPR_I32(byte offset); max
    // offset = 4*192*192*575*4B = 339 MB < 2^31, so i32 offsets are safe.
    // LDS dest from the real pointer: low 32 bits of a generic LDS address
    // are the wave-relative LDS byte offset, and the ptrtoint captures `sm`
    // so the "memory" clobber keeps the ds_load read-backs alive.
    auto issue = [&](int d, int bb) {
        const uint32_t sh2 = (uint32_t)(D_DISP - 1 - d);   // path-2 shear
        #pragma unroll
        for (int ch = 0; ch < 4; ++ch) {
            uint32_t g1 = (base + (uint32_t)ch * stride_ch + (uint32_t)d * stride_d) * 4u;
            uint32_t g2 = g1 + sh2 * 4u;
            uint32_t l1 = (uint32_t)(uintptr_t)(&sm[bb][ch][tid]);
            uint32_t l2 = (uint32_t)(uintptr_t)(&sm[bb][4 + ch][tid]);
            asm volatile("global_load_async_to_lds_b32 %0, %1, %2"
                         :: "v"(l1), "v"(g1), "s"(v) : "memory");
            asm volatile("global_load_async_to_lds_b32 %0, %1, %2"
                         :: "v"(l2), "v"(g2), "s"(v) : "memory");
        }
    };

    // Online-softmax state, path 1 and path 2.
    float m1 = -__builtin_inff(), s1 = 0.f, a1d = 0.f, a1c0 = 0.f, a1c1 = 0.f, a1c2 = 0.f;
    float m2 = -__builtin_inff(), s2 = 0.f, a2d = 0.f, a2c0 = 0.f, a2c1 = 0.f, a2c2 = 0.f;

    float t0, t1, t2, t3, t4, t5, t6, t7;     // current slice (load phase)

    auto loadSlice = [&](int bb) {
        t0 = sm[bb][0][tid]; t1 = sm[bb][1][tid];
        t2 = sm[bb][2][tid]; t3 = sm[bb][3][tid];
        t4 = sm[bb][4][tid]; t5 = sm[bb][5][tid];
        t6 = sm[bb][6][tid]; t7 = sm[bb][7][tid];
    };

    auto compute = [&](int d) {
        const float dv = (float)(D_DISP - 1 - d);

        float mn = fmaxf(m1, t0);
        float sc = __expf(m1 - mn);           // exp(-inf)=0 on first step
        float p  = __expf(t0 - mn);
        s1   = s1   * sc + p;
        a1d  = a1d  * sc + p * dv;
        a1c0 = a1c0 * sc + p * t1;
        a1c1 = a1c1 * sc + p * t2;
        a1c2 = a1c2 * sc + p * t3;
        m1 = mn;

        mn = fmaxf(m2, t4);
        sc = __expf(m2 - mn);
        p  = __expf(t4 - mn);
        s2   = s2   * sc + p;
        a2d  = a2d  * sc + p * dv;
        a2c0 = a2c0 * sc + p * t5;
        a2c1 = a2c1 * sc + p * t6;
        a2c2 = a2c2 * sc + p * t7;
        m2 = mn;
    };

    // Prologue: fill the 4-deep pipeline (32 async loads in flight per wave).
    #pragma unroll
    for (int p = 0; p < PIPE; ++p) issue(p, p);

    // Steady state: retire oldest batch, read it, WAR-fence, refill buffer.
    for (int d = 0; d < D_DISP - PIPE; ++d) {
        const int bb = d & (PIPE - 1);
        asm volatile("s_wait_asynccnt 24" ::: "memory");  // 8*(PIPE-1): slice d landed
        loadSlice(bb);
        asm volatile("s_wait_dscnt 0" ::: "memory");      // LDS reads retired before reuse
        issue(d + PIPE, bb);                              // overwrite freed buffer
        compute(d);
    }

    // Epilogue: last PIPE slices are fully in LDS after one drain.
    asm volatile("s_wait_asynccnt 0" ::: "memory");
    #pragma unroll
    for (int j = 0; j < PIPE; ++j) {
        const int d = D_DISP - PIPE + j;
        loadSlice(d & (PIPE - 1));
        compute(d);
    }

    const float inv1 = 1.0f / s1;
    const float inv2 = 1.0f / s2;
    const int HW  = H_DIM * W_DIM;
    const int pix = h * W_DIM + w;
    out[0 * HW + pix] = a1c0 * inv1;   // color_1, ch 0..2
    out[1 * HW + pix] = a1c1 * inv1;
    out[2 * HW + pix] = a1c2 * inv1;
    out[3 * HW + pix] = a2c0 * inv2;   // color_2, ch 0..2
    out[4 * HW + pix] = a2c1 * inv2;
    out[5 * HW + pix] = a2c2 * inv2;
    out[6 * HW + pix] = a1d * inv1;    // disp_1
    out[7 * HW + pix] = a2d * inv2;    // disp_2
}

extern "C" void kernel_launch(void* const* d_in, const int* in_sizes, int n_in,
                              void* d_out, int out_size, void* d_ws, size_t ws_size,
                              hipStream_t stream) {
    (void)in_sizes; (void)n_in; (void)out_size; (void)d_ws; (void)ws_size;
    const float* v = (const float*)d_in[0];   // x (d_in[1]) and y (d_in[2]) are shape-only
    float* out = (float*)d_out;
    dim3 grid(W_DIM / TPB, H_DIM, 1);         // (3, 192)
    volrender_kernel<<<grid, dim3(TPB, 1, 1), 0, stream>>>(v, out);
}